// CustomTSPInitEmbedding_52578989637700
// MI455X (gfx1250) — compile-verified
//
#include <hip/hip_runtime.h>
#include <hip/hip_bf16.h>

#define EMBED_DIM 128
#define KNN 10
#define NODE_DIM 22          // 2 + 2*K
#define FDIM 24              // padded to multiple of 4 for 16x16x4 WMMA
#define NNODES 2000          // == 125 * 16  -> tile-aligned tail
#define NBATCH 16
#define BLOCK 128
#define NODES_PER_BLOCK 128
#define BLOCKS_PER_BATCH 16  // 16*128 = 2048 >= 2000
#define FSTRIDE 25           // odd stride -> conflict-free LDS reads

typedef __attribute__((ext_vector_type(2))) float v2f;
typedef __attribute__((ext_vector_type(8))) float v8f;

__global__ __launch_bounds__(BLOCK)
void tsp_embed_kernel(const float* __restrict__ locs,   // (B, N, 2)
                      const float* __restrict__ W,      // (128, 22) row-major
                      const float* __restrict__ bias,   // (128,)
                      float* __restrict__ out)          // (B, N, 128)
{
    __shared__ float sLocs[2 * NNODES];                  // 16000 B
    __shared__ float sFeat[NODES_PER_BLOCK * FSTRIDE];   // 12800 B
    __shared__ float sW[FDIM * EMBED_DIM];               // 12288 B  (W^T, zero-padded)

    const int tid   = threadIdx.x;
    const int batch = blockIdx.x / BLOCKS_PER_BATCH;
    const int node0 = (blockIdx.x % BLOCKS_PER_BATCH) * NODES_PER_BLOCK;

    // ---- Phase 1: stage batch locs + padded W^T into LDS ----
    const float* bl = locs + (size_t)batch * NNODES * 2;
    for (int idx = tid; idx < 2 * NNODES; idx += BLOCK) sLocs[idx] = bl[idx];
    for (int idx = tid; idx < FDIM * EMBED_DIM; idx += BLOCK) {
        const int f = idx / EMBED_DIM;        // feature (padded K dim)
        const int e = idx % EMBED_DIM;        // embed column
        sW[idx] = (f < NODE_DIM) ? W[e * NODE_DIM + f] : 0.0f;
    }
    __syncthreads();

    // ---- Phase 2: per-thread KNN scan (squared distances; sqrt is monotone) ----
    const int node = node0 + tid;
    float bestD[KNN];
    int   bestJ[KNN];
#pragma unroll
    for (int k = 0; k < KNN; ++k) { bestD[k] = 3.0e38f; bestJ[k] = 0; }

    const float2* sl = (const float2*)sLocs;  // ds_load_b64 broadcast per candidate
    float px = 0.0f, py = 0.0f;
    if (node < NNODES) {
        const float2 me = sl[node];
        px = me.x; py = me.y;
        for (int j = 0; j < NNODES; ++j) {
            const float2 q = sl[j];
            const float dx = q.x - px;
            const float dy = q.y - py;
            const float d2 = dx * dx + dy * dy;
            if (j != node && d2 < bestD[KNN - 1]) {
                bestD[KNN - 1] = d2;
                bestJ[KNN - 1] = j;
#pragma unroll
                for (int s = KNN - 1; s > 0; --s) {
                    if (bestD[s] < bestD[s - 1]) {
                        const float td = bestD[s]; bestD[s] = bestD[s - 1]; bestD[s - 1] = td;
                        const int   tj = bestJ[s]; bestJ[s] = bestJ[s - 1]; bestJ[s - 1] = tj;
                    }
                }
            }
        }
    }

    // node features: [px, py, rel0.x, rel0.y, ..., rel9.x, rel9.y, 0, 0]
    float* fr = &sFeat[tid * FSTRIDE];
    if (node < NNODES) {
        fr[0] = px;
        fr[1] = py;
#pragma unroll
        for (int k = 0; k < KNN; ++k) {
            const float2 q = sl[bestJ[k]];
            fr[2 + 2 * k]     = q.x - px;     // rel = loc_j - loc_i
            fr[2 + 2 * k + 1] = q.y - py;
        }
        fr[NODE_DIM]     = 0.0f;              // pad
        fr[NODE_DIM + 1] = 0.0f;
    } else {
#pragma unroll
        for (int k = 0; k < FDIM; ++k) fr[k] = 0.0f;
    }
    __syncthreads();

    // ---- Phase 3: (128 nodes x 24) @ (24 x 128) via V_WMMA_F32_16X16X4_F32 ----
    // Each wave owns two 16-node tiles; B-fragments are shared between them.
    const int wave   = tid >> 5;              // wave32
    const int lane   = tid & 31;
    const int laneLo = lane & 15;
    const int kHalf  = (lane < 16) ? 0 : 2;   // K-half per ISA 16x4 f32 layout
    const int mBase  = (lane < 16) ? 0 : 8;   // C/D row half per ISA layout

    const int t0 = wave * 2;
    const int t1 = t0 + 1;
    // N is a multiple of 16, so tile validity is wave-uniform -> scalar guards,
    // EXEC stays all-ones at runtime around every WMMA.
    const bool ok0 = (node0 + t0 * 16) < NNODES;
    const bool ok1 = (node0 + t1 * 16) < NNODES;

    // A fragments for both tiles, all 6 K-chunks
    v2f a0[6], a1[6];
    {
        const float* fa0 = &sFeat[(t0 * 16 + laneLo) * FSTRIDE];
        const float* fa1 = &sFeat[(t1 * 16 + laneLo) * FSTRIDE];
#pragma unroll
        for (int kc = 0; kc < 6; ++kc) {
            const int k0 = kc * 4 + kHalf;
            a0[kc].x = fa0[k0];  a0[kc].y = fa0[k0 + 1];
            a1[kc].x = fa1[k0];  a1[kc].y = fa1[k0 + 1];
        }
    }

    float* const outB = out + (size_t)batch * NNODES * EMBED_DIM;

#pragma unroll
    for (int ct = 0; ct < 8; ++ct) {          // 8 column tiles of 16
        const int e = ct * 16 + laneLo;
        v2f bf[6];
#pragma unroll
        for (int kc = 0; kc < 6; ++kc) {      // B[k][n] from W^T, row striped
            const int k0 = kc * 4 + kHalf;
            bf[kc].x = sW[k0 * EMBED_DIM + e];
            bf[kc].y = sW[(k0 + 1) * EMBED_DIM + e];
        }
        v8f acc0 = {}, acc1 = {};
#pragma unroll
        for (int kc = 0; kc < 6; ++kc)
            acc0 = __builtin_amdgcn_wmma_f32_16x16x4_f32(
                false, a0[kc], false, bf[kc], (short)0, acc0, false, false);
#pragma unroll
        for (int kc = 0; kc < 6; ++kc)
            acc1 = __builtin_amdgcn_wmma_f32_16x16x4_f32(
                false, a1[kc], false, bf[kc], (short)0, acc1, false, false);

        const float bv = bias[e];
        if (ok0) {
            float* p = outB + (size_t)(node0 + t0 * 16 + mBase) * EMBED_DIM + e;
#pragma unroll
            for (int r = 0; r < 8; ++r)        // rows are +512B apart -> imm offsets
                p[(size_t)r * EMBED_DIM] = acc0[r] + bv;
        }
        if (ok1) {
            float* p = outB + (size_t)(node0 + t1 * 16 + mBase) * EMBED_DIM + e;
#pragma unroll
            for (int r = 0; r < 8; ++r)
                p[(size_t)r * EMBED_DIM] = acc1[r] + bv;
        }
    }
}

extern "C" void kernel_launch(void* const* d_in, const int* in_sizes, int n_in,
                              void* d_out, int out_size, void* d_ws, size_t ws_size,
                              hipStream_t stream) {
    const float* locs = (const float*)d_in[0];   // (16, 2000, 2)
    const float* W    = (const float*)d_in[1];   // (128, 22)
    const float* b    = (const float*)d_in[2];   // (128,)
    float* out        = (float*)d_out;           // (16, 2000, 128)

    dim3 grid(NBATCH * BLOCKS_PER_BATCH);        // 256 workgroups
    dim3 block(BLOCK);                           // 128 threads = 4 waves (wave32)
    tsp_embed_kernel<<<grid, block, 0, stream>>>(locs, W, b, out);
}